// Mamba3TileLang_55241869361709
// MI455X (gfx1250) — compile-verified
//
#include <hip/hip_runtime.h>
#include <hip/hip_bf16.h>

// ---------------------------------------------------------------------------
// Mamba3 chunked linear attention for MI455X (gfx1250, wave32, WMMA f32).
// Problem: b=2, L=4096, h=32, dv=64, dqk=128, chunk=64, nchunks=64.
// Bandwidth-bound (~300MB @ 23.3TB/s ~= 13us; 15 GFLOP is trivial), so we use
// full-precision V_WMMA_F32_16X16X4_F32 for all four GEMMs.
// ---------------------------------------------------------------------------

#define BATCH 2
#define LSEQ  4096
#define NH    32
#define DV    64
#define DQK   128
#define CCH   64          // chunk size
#define NCH   (LSEQ/CCH)  // 64 chunks
#define SQW   (DQK + 4)   // padded row stride for q/k tiles (bank-conflict pad)
#define SUW   (DV + 4)    // padded row stride for u / S / H tiles

typedef __attribute__((ext_vector_type(2))) float v2f;
typedef __attribute__((ext_vector_type(8))) float v8f;

__device__ __forceinline__ v8f wmma4(v2f a, v2f b, v8f c) {
  // D = A(16x4) * B(4x16) + C(16x16), fp32, wave32
  return __builtin_amdgcn_wmma_f32_16x16x4_f32(
      /*neg_a=*/false, a, /*neg_b=*/false, b,
      /*c_mod=*/(short)0, c, /*reuse_a=*/false, /*reuse_b=*/false);
}

// ---------------------------------------------------------------------------
// Kernel 0: per-(b,h) inclusive prefix sum of dt over the sequence -> gdt.
// gdt layout: [b][h][L] for coalesced per-(b,h) chunk reads later.
// ---------------------------------------------------------------------------
__global__ __launch_bounds__(32) void cumsum_dt_kernel(
    const float* __restrict__ dt, float* __restrict__ gdt) {
  const int h  = blockIdx.x % NH;
  const int bi = blockIdx.x / NH;
  const int lane = threadIdx.x;
  float carry = 0.f;
  float* grow = gdt + ((size_t)bi * NH + h) * LSEQ;
  for (int l0 = 0; l0 < LSEQ; l0 += 32) {
    float v = dt[((size_t)bi * LSEQ + l0 + lane) * NH + h];
    // wave32 inclusive scan
    for (int off = 1; off < 32; off <<= 1) {
      float nb = __shfl_up(v, off, 32);
      if (lane >= off) v += nb;
    }
    grow[l0 + lane] = carry + v;
    carry += __shfl(v, 31, 32);
  }
}

// ---------------------------------------------------------------------------
// Shared chunk builder: fills sQ = rot(q)*exp(cum), sK = rot(k)*exp(-cum),
// sU = dt*x for one (b, chunk, h). 128 threads: 2 rows per iteration.
// ---------------------------------------------------------------------------
__device__ __forceinline__ void build_chunk(
    int bi, int nc, int h,
    const float* __restrict__ xg, const float* __restrict__ dtg,
    const float* __restrict__ Ag, const float* __restrict__ Bgg,
    const float* __restrict__ Cgg, const float* __restrict__ gdt,
    float (*sQ)[SQW], float (*sK)[SQW], float (*sU)[SUW]) {
  const int tid  = threadIdx.x;
  const int l0   = nc * CCH;
  const float Ah = Ag[h];
  const float* gb = gdt + ((size_t)bi * NH + h) * LSEQ + l0;
  const float base = (nc == 0) ? 0.f : gb[-1];
  const int d    = tid & 63;
  const int toff = tid >> 6;
  for (int t0 = 0; t0 < CCH; t0 += 2) {
    const int t = t0 + toff;
    const size_t row = (size_t)bi * LSEQ + l0 + t;
    const float g   = gb[t];              // global cumsum(dt) -> rotary angle
    const float cum = Ah * (g - base);    // chunk-local decay cumsum
    const float eq  = __expf(cum);
    const float ek  = __expf(-cum);
    const float cc  = __cosf(g);
    const float sn  = __sinf(g);
    const float* Crow = Cgg + row * DQK;  // q (g=1 broadcast over heads)
    const float* Brow = Bgg + row * DQK;  // k
    const float q1 = Crow[d], q2 = Crow[d + 64];
    sQ[t][d]      = (q1 * cc - q2 * sn) * eq;
    sQ[t][d + 64] = (q1 * sn + q2 * cc) * eq;
    const float k1 = Brow[d], k2 = Brow[d + 64];
    sK[t][d]      = (k1 * cc - k2 * sn) * ek;
    sK[t][d + 64] = (k1 * sn + k2 * cc) * ek;
    const float dtv = dtg[row * NH + h];
    sU[t][d] = dtv * xg[(row * NH + h) * DV + d];
  }
}

// ---------------------------------------------------------------------------
// Kernel 1: per-(b, chunk, h) intra-chunk path.
//   S = tril(qt . ktT)   (64x64, K=128)
//   Yintra = S . U       (64x64, K=64)
//   out = 4*Yintra + D*x
// 128 threads = 4 waves; wave w owns rows [16w,16w+16) of the 64x64 output.
// ---------------------------------------------------------------------------
__global__ __launch_bounds__(128) void chunk_intra_kernel(
    const float* __restrict__ xg, const float* __restrict__ dtg,
    const float* __restrict__ Ag, const float* __restrict__ Bgg,
    const float* __restrict__ Cgg, const float* __restrict__ Dg,
    const float* __restrict__ gdt, float* __restrict__ out) {
  const int blk = blockIdx.x;
  const int h   = blk % NH;
  const int nc  = (blk / NH) % NCH;
  const int bi  = blk / (NH * NCH);

  extern __shared__ float smem[];
  float (*sQ)[SQW] = (float (*)[SQW])(smem);
  float (*sK)[SQW] = (float (*)[SQW])(smem + CCH * SQW);
  float (*sU)[SUW] = (float (*)[SUW])(smem + 2 * CCH * SQW);
  float (*sS)[SUW] = (float (*)[SUW])(smem + 2 * CCH * SQW + CCH * SUW);

  build_chunk(bi, nc, h, xg, dtg, Ag, Bgg, Cgg, gdt, sQ, sK, sU);
  __syncthreads();

  const int tid  = threadIdx.x;
  const int wave = tid >> 5;
  const int lane = tid & 31;
  const int half = lane >> 4;
  const int l16  = lane & 15;
  const int m0   = wave * 16;

  // --- GEMM1: S = qt * ktT ---
  v8f accS[4];
  for (int nt = 0; nt < 4; ++nt) accS[nt] = (v8f){0,0,0,0,0,0,0,0};
  for (int k0 = 0; k0 < DQK; k0 += 4) {
    const int kk = k0 + half * 2;
    v2f a; a.x = sQ[m0 + l16][kk]; a.y = sQ[m0 + l16][kk + 1];
    for (int nt = 0; nt < 4; ++nt) {
      const int col = nt * 16 + l16;          // B[k][n] = ktT[k][n] = sK[n][k]
      v2f b; b.x = sK[col][kk]; b.y = sK[col][kk + 1];
      accS[nt] = wmma4(a, b, accS[nt]);
    }
  }
  // tril mask (decay factors already folded into qt/kt scaling) -> sS
  for (int nt = 0; nt < 4; ++nt)
    for (int r = 0; r < 8; ++r) {
      const int M = m0 + half * 8 + r;        // query index t
      const int N = nt * 16 + l16;            // key index s
      sS[M][N] = (M >= N) ? accS[nt][r] : 0.f;
    }
  __syncthreads();

  // --- GEMM2: Yintra = S * U ---
  v8f accY[4];
  for (int nt = 0; nt < 4; ++nt) accY[nt] = (v8f){0,0,0,0,0,0,0,0};
  for (int k0 = 0; k0 < CCH; k0 += 4) {
    const int kk = k0 + half * 2;
    v2f a; a.x = sS[m0 + l16][kk]; a.y = sS[m0 + l16][kk + 1];
    for (int nt = 0; nt < 4; ++nt) {
      const int col = nt * 16 + l16;
      v2f b; b.x = sU[kk][col]; b.y = sU[kk + 1][col];
      accY[nt] = wmma4(a, b, accY[nt]);
    }
  }
  // out = MIMO_RANK * y_intra + D*x   (y_inter added by the scan kernel)
  const float dvh = Dg[h];
  const int l0 = nc * CCH;
  for (int nt = 0; nt < 4; ++nt)
    for (int r = 0; r < 8; ++r) {
      const int M = m0 + half * 8 + r;
      const int e = nt * 16 + l16;
      const size_t idx = (((size_t)bi * LSEQ + l0 + M) * NH + h) * DV + e;
      out[idx] = 4.f * accY[nt][r] + dvh * xg[idx];
    }
}

// ---------------------------------------------------------------------------
// Kernel 2: per-(b,h) sequential scan over chunks with H (128x64) in LDS.
//   Yinter = qt . Hprev     (64x64, K=128)   -> out += 4*Yinter
//   dH     = ktT . U        (128x64, K=64)
//   H      = exp(cum_last)*H + dH
// ---------------------------------------------------------------------------
__global__ __launch_bounds__(128) void scan_inter_kernel(
    const float* __restrict__ xg, const float* __restrict__ dtg,
    const float* __restrict__ Ag, const float* __restrict__ Bgg,
    const float* __restrict__ Cgg, const float* __restrict__ gdt,
    float* __restrict__ out) {
  const int h  = blockIdx.x % NH;
  const int bi = blockIdx.x / NH;

  extern __shared__ float smem[];
  float (*sQ)[SQW] = (float (*)[SQW])(smem);
  float (*sK)[SQW] = (float (*)[SQW])(smem + CCH * SQW);
  float (*sU)[SUW] = (float (*)[SUW])(smem + 2 * CCH * SQW);
  float (*sH)[SUW] = (float (*)[SUW])(smem + 2 * CCH * SQW + CCH * SUW);

  const int tid  = threadIdx.x;
  const int wave = tid >> 5;
  const int lane = tid & 31;
  const int half = lane >> 4;
  const int l16  = lane & 15;
  const int m0   = wave * 16;
  const float Ah = Ag[h];

  // H0 = 0
  for (int i = tid; i < DQK * SUW; i += 128) ((float*)sH)[i] = 0.f;
  __syncthreads();

  for (int nc = 0; nc < NCH; ++nc) {
    build_chunk(bi, nc, h, xg, dtg, Ag, Bgg, Cgg, gdt, sQ, sK, sU);
    __syncthreads();

    // --- Yinter = qt * Hprev ---
    v8f accY[4];
    for (int nt = 0; nt < 4; ++nt) accY[nt] = (v8f){0,0,0,0,0,0,0,0};
    for (int k0 = 0; k0 < DQK; k0 += 4) {
      const int kk = k0 + half * 2;
      v2f a; a.x = sQ[m0 + l16][kk]; a.y = sQ[m0 + l16][kk + 1];
      for (int nt = 0; nt < 4; ++nt) {
        const int col = nt * 16 + l16;
        v2f b; b.x = sH[kk][col]; b.y = sH[kk + 1][col];
        accY[nt] = wmma4(a, b, accY[nt]);
      }
    }

    // --- dH = ktT * U : wave owns d-rows [32w, 32w+32) -> 2x4 tiles ---
    v8f accH[2][4];
    for (int mt = 0; mt < 2; ++mt)
      for (int nt = 0; nt < 4; ++nt) accH[mt][nt] = (v8f){0,0,0,0,0,0,0,0};
    for (int k0 = 0; k0 < CCH; k0 += 4) {
      const int kk = k0 + half * 2;          // k = s (time within chunk)
      for (int mt = 0; mt < 2; ++mt) {
        const int Mrow = wave * 32 + mt * 16 + l16;   // m = d index
        v2f a; a.x = sK[kk][Mrow]; a.y = sK[kk + 1][Mrow]; // A[m][k]=sK[k][m]
        for (int nt = 0; nt < 4; ++nt) {
          const int col = nt * 16 + l16;
          v2f b; b.x = sU[kk][col]; b.y = sU[kk + 1][col];
          accH[mt][nt] = wmma4(a, b, accH[mt][nt]);
        }
      }
    }
    __syncthreads();   // all reads of sH / sK / sU complete

    // out += 4 * Yinter
    const int l0 = nc * CCH;
    for (int nt = 0; nt < 4; ++nt)
      for (int r = 0; r < 8; ++r) {
        const int M = m0 + half * 8 + r;
        const int e = nt * 16 + l16;
        const size_t idx = (((size_t)bi * LSEQ + l0 + M) * NH + h) * DV + e;
        out[idx] += 4.f * accY[nt][r];
      }

    // H = cdecay * H + dH
    const float* gb = gdt + ((size_t)bi * NH + h) * LSEQ + l0;
    const float base = (nc == 0) ? 0.f : gb[-1];
    const float cdec = __expf(Ah * (gb[CCH - 1] - base));
    for (int mt = 0; mt < 2; ++mt)
      for (int nt = 0; nt < 4; ++nt)
        for (int r = 0; r < 8; ++r) {
          const int drow = wave * 32 + mt * 16 + half * 8 + r;
          const int e    = nt * 16 + l16;
          sH[drow][e] = cdec * sH[drow][e] + accH[mt][nt][r];
        }
    __syncthreads();
  }
}

// ---------------------------------------------------------------------------
extern "C" void kernel_launch(void* const* d_in, const int* in_sizes, int n_in,
                              void* d_out, int out_size, void* d_ws, size_t ws_size,
                              hipStream_t stream) {
  (void)in_sizes; (void)n_in; (void)out_size; (void)ws_size;
  const float* x  = (const float*)d_in[0];
  const float* dt = (const float*)d_in[1];
  const float* A  = (const float*)d_in[2];
  const float* Bg = (const float*)d_in[3];
  const float* Cg = (const float*)d_in[4];
  const float* D  = (const float*)d_in[5];
  float* out = (float*)d_out;
  float* gdt = (float*)d_ws;   // [b][h][L] floats = 1 MB

  cumsum_dt_kernel<<<BATCH * NH, 32, 0, stream>>>(dt, gdt);

  const size_t smemB = (size_t)(2 * CCH * SQW + 2 * CCH * SUW) * sizeof(float);
  chunk_intra_kernel<<<BATCH * NCH * NH, 128, smemB, stream>>>(
      x, dt, A, Bg, Cg, D, gdt, out);

  const size_t smemC = (size_t)(2 * CCH * SQW + CCH * SUW + DQK * SUW) * sizeof(float);
  scan_inter_kernel<<<BATCH * NH, 128, smemC, stream>>>(
      x, dt, A, Bg, Cg, gdt, out);
}